// CapsuleSubLayer_20555713478761
// MI455X (gfx1250) — compile-verified
//
#include <hip/hip_runtime.h>
#include <hip/hip_bf16.h>

typedef __attribute__((ext_vector_type(2))) float v2f;
typedef __attribute__((ext_vector_type(8))) float v8f;

#define JB 8192      // bsz*seq
#define NJ 8         // num_out
#define DIM 128      // in/out dim
#define COLS 1024    // NJ*DIM

__device__ __forceinline__ float squash_gain(float n2) {
    // (n2/(1+n2)) / sqrt(n2+eps)
    return n2 / ((1.0f + n2) * sqrtf(n2 + 1e-8f));
}

// ---------------------------------------------------------------------------
// Kernel 1: uh[r, j*128+e] = sum_d x[7, b, s, d] * W[7, j, d, e],  r = s*8+b
// fp32 WMMA 16x16x4. One wave = 16 rows; 8 waves/block = 128 rows; 64 blocks.
// ---------------------------------------------------------------------------
__launch_bounds__(256)
__global__ void gemm_uhat_kernel(const float* __restrict__ x,
                                 const float* __restrict__ w,
                                 float* __restrict__ uh) {
    const float* x7 = x + (size_t)7 * 8 * 1024 * 128;   // (bsz, seq, dim)
    const float* w7 = w + (size_t)7 * 8 * 128 * 128;    // (j, d, e)

    const int wave = threadIdx.x >> 5;
    const int lane = threadIdx.x & 31;
    const int half = lane >> 4;       // 0: K+{0,1} / rows M..M+7 ; 1: K+{2,3} / rows M+8..
    const int l16  = lane & 15;

    const int row0 = blockIdx.x * 128 + wave * 16;      // M-tile base (r index)
    const int r    = row0 + l16;                        // A row for this lane
    // r = s*8 + b  ->  x7 offset ((b)*1024 + s)*128
    const float* arow = x7 + (size_t)((r & 7) * 1024 + (r >> 3)) * 128;

    for (int j = 0; j < NJ; ++j) {
        const float* wj = w7 + (size_t)j * 128 * 128;
        v8f acc[8];
#pragma unroll
        for (int n = 0; n < 8; ++n) acc[n] = (v8f){0,0,0,0,0,0,0,0};

        for (int k = 0; k < 128; k += 4) {
            // A fragment: lane holds X[r, k+2*half] , X[r, k+2*half+1]
            v2f a;
            a.x = arow[k + 2 * half + 0];
            a.y = arow[k + 2 * half + 1];
            const float* wk = wj + (size_t)(k + 2 * half) * 128 + l16;
#pragma unroll
            for (int n = 0; n < 8; ++n) {
                v2f b;
                b.x = wk[n * 16];          // W[j, k+2h,   n0+l16]
                b.y = wk[128 + n * 16];    // W[j, k+2h+1, n0+l16]
                acc[n] = __builtin_amdgcn_wmma_f32_16x16x4_f32(
                    false, a, false, b, (short)0, acc[n], false, false);
            }
        }
        // store D: vgpr g -> row row0 + g + 8*half, col n*16 + l16
#pragma unroll
        for (int n = 0; n < 8; ++n) {
            float* od = uh + (size_t)(row0 + 8 * half) * COLS + j * DIM + n * 16 + l16;
#pragma unroll
            for (int g = 0; g < 8; ++g) {
                od[(size_t)g * COLS] = acc[n][g];
            }
        }
    }
}

// ---------------------------------------------------------------------------
// Kernel 2: column means  uhm[c] = mean_r uh[r, c]   (c = j*128+e, 1024 cols)
// ---------------------------------------------------------------------------
__launch_bounds__(256)
__global__ void colmean_kernel(const float* __restrict__ uh, float* __restrict__ uhm) {
    int c = blockIdx.x * 256 + threadIdx.x;   // 4 blocks x 256 = 1024
    float s = 0.0f;
    for (int r = 0; r < JB; ++r) s += uh[(size_t)r * COLS + c];
    uhm[c] = s * (1.0f / (float)JB);
}

// ---------------------------------------------------------------------------
// Kernel 3: per (r,j): dotp = uh.uhm , n2 = |uh|^2. One wave per pair.
// ---------------------------------------------------------------------------
__launch_bounds__(256)
__global__ void dotn2_kernel(const float* __restrict__ uh,
                             const float* __restrict__ uhm,
                             float* __restrict__ dotp, float* __restrict__ n2) {
    const int wave = threadIdx.x >> 5;
    const int lane = threadIdx.x & 31;
    const int pair = blockIdx.x * 8 + wave;   // 8192 blocks -> 65536 pairs
    const int r = pair >> 3, j = pair & 7;

    const float4 a = ((const float4*)(uh + (size_t)r * COLS + j * DIM))[lane];
    const float4 m = ((const float4*)(uhm + j * DIM))[lane];
    float d = a.x * m.x + a.y * m.y + a.z * m.z + a.w * m.w;
    float q = a.x * a.x + a.y * a.y + a.z * a.z + a.w * a.w;
#pragma unroll
    for (int off = 16; off > 0; off >>= 1) {
        d += __shfl_xor(d, off, 32);
        q += __shfl_xor(q, off, 32);
    }
    if (lane == 0) { dotp[pair] = d; n2[pair] = q; }
}

// ---------------------------------------------------------------------------
// Kernel 4: routing (2 B-updates), writes final coupling coeffs cf[8].
// Single block, 1024 threads.
// ---------------------------------------------------------------------------
__launch_bounds__(1024)
__global__ void routing_kernel(const float* __restrict__ dotp,
                               const float* __restrict__ n2,
                               float* __restrict__ cf) {
    __shared__ float B7[8];
    __shared__ float c[8];
    __shared__ float sumj[8];
    const int tid = threadIdx.x;
    if (tid < 8) B7[tid] = 0.0f;

    for (int it = 0; it < 2; ++it) {
        __syncthreads();
        if (tid == 0) {
            float m = B7[0];
            for (int jj = 1; jj < 8; ++jj) m = fmaxf(m, B7[jj]);
            float s = 0.0f, e[8];
            for (int jj = 0; jj < 8; ++jj) { e[jj] = __expf(B7[jj] - m); s += e[jj]; }
            for (int jj = 0; jj < 8; ++jj) c[jj] = e[jj] / s;
        }
        __syncthreads();
        if (tid < 8) sumj[tid] = 0.0f;
        __syncthreads();
        // stride 1024 is a multiple of 8 -> each thread touches one fixed j
        const int j = tid & 7;
        const float cj = c[j];
        float a = 0.0f;
        for (int p = tid; p < JB * NJ; p += 1024)
            a += squash_gain(cj * cj * n2[p]) * dotp[p];
        atomicAdd(&sumj[j], a);
        __syncthreads();
        if (tid < 8) B7[tid] += c[tid] * sumj[tid] * (1.0f / (float)JB);
    }
    __syncthreads();
    if (tid == 0) {
        float m = B7[0];
        for (int jj = 1; jj < 8; ++jj) m = fmaxf(m, B7[jj]);
        float s = 0.0f, e[8];
        for (int jj = 0; jj < 8; ++jj) { e[jj] = __expf(B7[jj] - m); s += e[jj]; }
        for (int jj = 0; jj < 8; ++jj) cf[jj] = e[jj] / s;
    }
}

// ---------------------------------------------------------------------------
// Kernel 5: out[r, j*128+e] = cf[j] * g(cf[j]^2 * n2[r,j]) * uh[r, j*128+e]
// ---------------------------------------------------------------------------
__launch_bounds__(256)
__global__ void final_kernel(const float* __restrict__ uh,
                             const float* __restrict__ n2,
                             const float* __restrict__ cf,
                             float* __restrict__ out) {
    const size_t idx  = (size_t)blockIdx.x * 256 + threadIdx.x;  // float4 index
    const size_t base = idx * 4;                                  // element index
    const int r = (int)(base >> 10);
    const int j = (int)((base >> 7) & 7);
    const float cj = cf[j];
    const float gain = cj * squash_gain(cj * cj * n2[r * 8 + j]);
    float4 a = *(const float4*)(uh + base);
    a.x *= gain; a.y *= gain; a.z *= gain; a.w *= gain;
    *(float4*)(out + base) = a;
}

// ---------------------------------------------------------------------------
extern "C" void kernel_launch(void* const* d_in, const int* in_sizes, int n_in,
                              void* d_out, int out_size, void* d_ws, size_t ws_size,
                              hipStream_t stream) {
    const float* x = (const float*)d_in[0];   // (8, 8, 1024, 128)
    const float* w = (const float*)d_in[1];   // (8, 8, 128, 128)
    float* out = (float*)d_out;               // (8192, 1024)

    char* ws = (char*)d_ws;
    float* uh   = (float*)(ws);                                    // 8192*1024 f32 = 32 MB
    float* n2   = (float*)(ws + (size_t)JB * COLS * 4);            // 65536 f32
    float* dotp = (float*)(ws + (size_t)JB * COLS * 4 + 262144);   // 65536 f32
    float* uhm  = (float*)(ws + (size_t)JB * COLS * 4 + 524288);   // 1024 f32
    float* cf   = (float*)(ws + (size_t)JB * COLS * 4 + 528384);   // 8 f32

    gemm_uhat_kernel<<<64, 256, 0, stream>>>(x, w, uh);
    colmean_kernel<<<4, 256, 0, stream>>>(uh, uhm);
    dotn2_kernel<<<JB, 256, 0, stream>>>(uh, uhm, dotp, n2);
    routing_kernel<<<1, 1024, 0, stream>>>(dotp, n2, cf);
    final_kernel<<<(JB * COLS / 4) / 256, 256, 0, stream>>>(uh, n2, cf, out);
}